// Hete_GCN_layers_2834678415702
// MI455X (gfx1250) — compile-verified
//
#include <hip/hip_runtime.h>
#include <math.h>

#define D 128

typedef __attribute__((ext_vector_type(2))) float v2f;
typedef __attribute__((ext_vector_type(8))) float v8f;

// ---------------- CSR build ----------------

__global__ void zero_i32(int* __restrict__ p, int n) {
  int i = blockIdx.x * blockDim.x + threadIdx.x;
  if (i < n) p[i] = 0;
}

__global__ void count_deg(const int* __restrict__ row, int* __restrict__ deg, int e) {
  int i = blockIdx.x * blockDim.x + threadIdx.x;
  if (i < e) atomicAdd(&deg[row[i]], 1);
}

// Single-workgroup chunked exclusive scan (runs once per call; N=100K -> ~98 chunks).
__global__ __launch_bounds__(1024) void scan_deg(const int* __restrict__ deg,
                                                 int* __restrict__ row_ptr,
                                                 int* __restrict__ cursor, int n) {
  __shared__ int lds[1024];
  const int t = threadIdx.x;
  int carry = 0;
  for (int base = 0; base < n; base += 1024) {
    int i = base + t;
    int v = (i < n) ? deg[i] : 0;
    lds[t] = v;
    __syncthreads();
    for (int off = 1; off < 1024; off <<= 1) {
      int add = (t >= off) ? lds[t - off] : 0;
      __syncthreads();
      lds[t] += add;
      __syncthreads();
    }
    int excl = carry + lds[t] - v;
    if (i < n) { row_ptr[i] = excl; cursor[i] = excl; }
    int tot = lds[1023];
    __syncthreads();
    carry += tot;
  }
  if (t == 0) row_ptr[n] = carry;
}

__global__ void fill_csr(const int* __restrict__ row, const int* __restrict__ col,
                         const float* __restrict__ val, int* __restrict__ cursor,
                         int* __restrict__ csr_col, float* __restrict__ csr_val, int e) {
  int i = blockIdx.x * blockDim.x + threadIdx.x;
  if (i < e) {
    int p = atomicAdd(&cursor[row[i]], 1);
    csr_col[p] = col[i];
    csr_val[p] = val[i];
  }
}

// ---------------- normalization ----------------

__global__ void calc_dinv(const int* __restrict__ row_ptr, const float* __restrict__ csr_val,
                          float* __restrict__ dinv, int n) {
  int r = blockIdx.x * blockDim.x + threadIdx.x;
  if (r >= n) return;
  int s0 = row_ptr[r], s1 = row_ptr[r + 1];
  float s = 0.f;
  for (int i = s0; i < s1; ++i) s += csr_val[i];
  dinv[r] = (s > 0.f) ? (1.0f / sqrtf(s)) : 0.0f;
}

__global__ void norm_vals(const int* __restrict__ row_ptr, const int* __restrict__ csr_col,
                          float* __restrict__ csr_val, const float* __restrict__ dinv, int n) {
  int r = blockIdx.x * blockDim.x + threadIdx.x;
  if (r >= n) return;
  int s0 = row_ptr[r], s1 = row_ptr[r + 1];
  float dr = dinv[r];
  for (int i = s0; i < s1; ++i)
    csr_val[i] = csr_val[i] * dr * dinv[csr_col[i]];
}

__global__ void softmax3(const float* __restrict__ a_in, float* __restrict__ a3) {
  float x0 = a_in[0], x1 = a_in[1], x2 = a_in[2];
  float m = fmaxf(x0, fmaxf(x1, x2));
  float e0 = expf(x0 - m), e1 = expf(x1 - m), e2 = expf(x2 - m);
  float s = e0 + e1 + e2;
  a3[0] = e0 / s; a3[1] = e1 / s; a3[2] = e2 / s;
}

// ---------------- SpMM: one wave32 per output row, fused permutation scatter ---------
// accumulate==0: out[index[r]] = acc        (layer 1 -> h1)
// accumulate==1: out[index[r]] += a3[2]*acc (layer 2 fused into final result)

__global__ void spmm_scatter(const float* __restrict__ hsrc, const int* __restrict__ row_ptr,
                             const int* __restrict__ csr_col, const float* __restrict__ csr_val,
                             const int* __restrict__ index, const float* __restrict__ a3,
                             float* __restrict__ out, int n, int accumulate) {
  int gid = blockIdx.x * blockDim.x + threadIdx.x;
  int wv = gid >> 5;            // wave id == row id
  int lane = gid & 31;
  if (wv >= n) return;
  int s0 = row_ptr[wv], s1 = row_ptr[wv + 1];
  const float* hl = hsrc + (size_t)lane * 4;   // lane covers 4 of the 128 features
  float4 acc = make_float4(0.f, 0.f, 0.f, 0.f);
  int c = (s0 < s1) ? csr_col[s0] : 0;
  for (int s = s0; s < s1; ++s) {
    int cn = (s + 1 < s1) ? csr_col[s + 1] : c;
    __builtin_prefetch(hl + (size_t)cn * D, 0, 3);   // global_prefetch_b8: next gathered row
    float v = csr_val[s];
    float4 hv = *(const float4*)(hl + (size_t)c * D);
    acc.x = fmaf(v, hv.x, acc.x);
    acc.y = fmaf(v, hv.y, acc.y);
    acc.z = fmaf(v, hv.z, acc.z);
    acc.w = fmaf(v, hv.w, acc.w);
    c = cn;
  }
  int p = index[wv];                                  // permutation scatter target
  float* op = out + (size_t)p * D + lane * 4;
  if (accumulate) {
    float a2 = a3[2];
    float4 cur = *(const float4*)op;
    cur.x = fmaf(a2, acc.x, cur.x);
    cur.y = fmaf(a2, acc.y, cur.y);
    cur.z = fmaf(a2, acc.z, cur.z);
    cur.w = fmaf(a2, acc.w, cur.w);
    *(float4*)op = cur;
  } else {
    *(float4*)op = acc;
  }
}

// ---------------- WMMA combine: out = a0*F + a1*H1 (exact, per 16x16 tile) -----------
// For each 4-col chunk c: A_c(16x4) x (a * E_c)(4x16) sums to a*Tile, with E_c the
// shifted identity. B built in registers per the ISA 4x16 layout (lane halves hold
// K={0,1} / K={2,3} in VGPR0/VGPR1, column = lane%16).

__global__ void combine_wmma(const float* __restrict__ F, const float* __restrict__ H1,
                             const float* __restrict__ a3, float* __restrict__ out, int n) {
  int gid = blockIdx.x * blockDim.x + threadIdx.x;
  int wv = gid >> 5;
  int lane = gid & 31;
  int mg = wv >> 3;         // 16-row group
  int tn = wv & 7;          // 16-col tile within D=128
  int lm = lane & 15;
  int hi = lane >> 4;
  float a0 = a3[0], a1 = a3[1];

  int row = mg * 16 + lm;
  if (row >= n) row = n - 1;               // clamp: keeps EXEC all-ones for WMMA
  size_t rbase = (size_t)row * D + (size_t)tn * 16 + (size_t)hi * 2;

  v8f acc = {};
#pragma unroll
  for (int cch = 0; cch < 4; ++cch) {      // a0 * F tile
    v2f A = *(const v2f*)(F + rbase + cch * 4);
    v2f B;
    B.x = (lm == cch * 4 + hi * 2)     ? a0 : 0.f;
    B.y = (lm == cch * 4 + hi * 2 + 1) ? a0 : 0.f;
    acc = __builtin_amdgcn_wmma_f32_16x16x4_f32(false, A, false, B, (short)0, acc,
                                                false, false);
  }
#pragma unroll
  for (int cch = 0; cch < 4; ++cch) {      // + a1 * H1 tile
    v2f A = *(const v2f*)(H1 + rbase + cch * 4);
    v2f B;
    B.x = (lm == cch * 4 + hi * 2)     ? a1 : 0.f;
    B.y = (lm == cch * 4 + hi * 2 + 1) ? a1 : 0.f;
    acc = __builtin_amdgcn_wmma_f32_16x16x4_f32(false, A, false, B, (short)0, acc,
                                                false, false);
  }
#pragma unroll
  for (int v = 0; v < 8; ++v) {            // C/D layout: row = v + 8*hi, col = lm
    int orow = mg * 16 + v + hi * 8;
    if (orow < n) out[(size_t)orow * D + (size_t)tn * 16 + lm] = acc[v];
  }
}

// ---------------- launch ----------------

extern "C" void kernel_launch(void* const* d_in, const int* in_sizes, int n_in,
                              void* d_out, int out_size, void* d_ws, size_t ws_size,
                              hipStream_t stream) {
  const float* features = (const float*)d_in[0];
  const int*   edge_row = (const int*)d_in[1];
  const int*   edge_col = (const int*)d_in[2];
  const float* edge_val = (const float*)d_in[3];
  const int*   index    = (const int*)d_in[4];
  const float* a_in     = (const float*)d_in[5];
  float* out = (float*)d_out;

  const int n = in_sizes[0] / D;
  const int e = in_sizes[1];
  (void)n_in; (void)out_size; (void)ws_size;

  char* w = (char*)d_ws;
  size_t off = 0;
  auto take = [&](size_t bytes) -> char* {
    char* p = w + off;
    off += (bytes + 255) & ~(size_t)255;
    return p;
  };
  int*   deg     = (int*)take((size_t)n * 4);
  int*   row_ptr = (int*)take(((size_t)n + 1) * 4);
  int*   cursor  = (int*)take((size_t)n * 4);
  int*   csr_col = (int*)take((size_t)e * 4);
  float* csr_val = (float*)take((size_t)e * 4);
  float* dinv    = (float*)take((size_t)n * 4);
  float* h1      = (float*)take((size_t)n * D * 4);
  float* a3      = (float*)take(4 * 4);

  const int tb = 256;
  zero_i32 <<<(n + tb - 1) / tb, tb, 0, stream>>>(deg, n);
  count_deg<<<(e + tb - 1) / tb, tb, 0, stream>>>(edge_row, deg, e);
  scan_deg <<<1, 1024, 0, stream>>>(deg, row_ptr, cursor, n);
  fill_csr <<<(e + tb - 1) / tb, tb, 0, stream>>>(edge_row, edge_col, edge_val,
                                                  cursor, csr_col, csr_val, e);
  calc_dinv<<<(n + tb - 1) / tb, tb, 0, stream>>>(row_ptr, csr_val, dinv, n);
  norm_vals<<<(n + tb - 1) / tb, tb, 0, stream>>>(row_ptr, csr_col, csr_val, dinv, n);
  softmax3 <<<1, 1, 0, stream>>>(a_in, a3);

  // Layer 1: h1[index[r]] = sum_slots val * F[col]
  long long thr = (long long)n * 32;
  spmm_scatter<<<(int)((thr + tb - 1) / tb), tb, 0, stream>>>(
      features, row_ptr, csr_col, csr_val, index, a3, h1, n, 0);

  // out = a0*F + a1*h1 via v_wmma_f32_16x16x4_f32
  long long ctiles = ((long long)(n + 15) / 16) * 8;
  long long cthr = ctiles * 32;
  combine_wmma<<<(int)((cthr + tb - 1) / tb), tb, 0, stream>>>(features, h1, a3, out, n);

  // Layer 2 fused: out[index[r]] += a2 * (sum_slots val * h1[col])
  spmm_scatter<<<(int)((thr + tb - 1) / tb), tb, 0, stream>>>(
      h1, row_ptr, csr_col, csr_val, index, a3, out, n, 1);
}